// TimeConvIncpetionBlock_35957466202492
// MI455X (gfx1250) — compile-verified
//
#include <hip/hip_runtime.h>
#include <cmath>

// CDNA5 / gfx1250, wave32.
typedef float v2f __attribute__((ext_vector_type(2)));
typedef float v8f __attribute__((ext_vector_type(8)));

#define B_   32
#define C_   256
#define T_   4096
#define TB   32               // timesteps per block (2 x 16-wide WMMA subtiles)
#define OB   128              // output channels per block (8 waves x 16 rows)
#define HALO 2                // k=5 depthwise needs +/-2
#define ATW  (TB + 2 * HALO)  // staged tile width = 36

__device__ __forceinline__ float gelu_exact(float v) {
    // torch default GELU: 0.5*x*(1+erf(x/sqrt(2)))
    return 0.5f * v * (1.0f + erff(v * 0.70710678118654752440f));
}

__global__ __launch_bounds__(256)
void timeconv_inception_kernel(const float* __restrict__ x,
                               const float* __restrict__ w1,
                               const float* __restrict__ b1,
                               const float* __restrict__ w3,
                               const float* __restrict__ b3,
                               const float* __restrict__ w5,
                               const float* __restrict__ b5,
                               const float* __restrict__ wc,
                               const float* __restrict__ bc,
                               float* __restrict__ out) {
    __shared__ float aT[C_ * ATW];       // gelu(x): all 256 in-channels x 36 (36.9 KB)
    __shared__ float cb1[OB], cb3[OB], cb5[OB], cbc[OB];
    __shared__ float cw3[OB * 3], cw5[OB * 5], cwc[OB * 3];

    const int tid = threadIdx.x;
    const int tt0 = blockIdx.x * TB;   // timestep tile origin
    const int ob0 = blockIdx.y * OB;   // output-channel block origin
    const int bb  = blockIdx.z;        // batch

    // ---- Stage gelu(x) for all 256 channels, width TB + halo (zero-pad OOB t) ----
    const float* xb = x + (size_t)bb * C_ * T_;
    for (int i = tid; i < C_ * ATW; i += 256) {
        const int c  = i / ATW;
        const int u  = i - c * ATW;
        const int tg = tt0 - HALO + u;
        float v = (tg >= 0 && tg < T_) ? xb[c * T_ + tg] : 0.0f;
        aT[i] = gelu_exact(v);
    }
    // ---- Stage per-channel coefficients for this block's 128 output channels ----
    if (tid < OB) {
        const int o = ob0 + tid;
        cb1[tid] = b1[o];
        cb3[tid] = b3[o];
        cb5[tid] = b5[o];
        cbc[tid] = bc[o];
        cw3[tid * 3 + 0] = w3[o * 3 + 0];
        cw3[tid * 3 + 1] = w3[o * 3 + 1];
        cw3[tid * 3 + 2] = w3[o * 3 + 2];
        cw5[tid * 5 + 0] = w5[o * 5 + 0];
        cw5[tid * 5 + 1] = w5[o * 5 + 1];
        cw5[tid * 5 + 2] = w5[o * 5 + 2];
        cw5[tid * 5 + 3] = w5[o * 5 + 3];
        cw5[tid * 5 + 4] = w5[o * 5 + 4];
        cwc[tid * 3 + 0] = wc[o * 3 + 0];
        cwc[tid * 3 + 1] = wc[o * 3 + 1];
        cwc[tid * 3 + 2] = wc[o * 3 + 2];
    }
    __syncthreads();

    // ---- Per-wave: 16 o-rows x 32 t-cols = two 16x16 tiles sharing A fragments ----
    const int lane = tid & 31;
    const int w    = tid >> 5;          // wave id 0..7
    const int half = lane >> 4;         // 0: lanes 0-15, 1: lanes 16-31
    const int l15  = lane & 15;
    const int oT   = ob0 + 16 * w;      // this wave's 16-row o origin (absolute)

    // A fragment (w1, 16x4 f32): lanes 0-15 -> K={0,1}, lanes 16-31 -> K={2,3}
    const float* ap  = w1 + (oT + l15) * C_ + 2 * half;
    // B fragments (a, 4x16 f32): same K split, N = t across lanes; two t-subtiles
    const float* bp0 = &aT[(2 * half) * ATW + l15 + HALO];        // t-subtile 0
    const float* bp1 = bp0 + 16;                                  // t-subtile 1

    v8f acc0 = {0.f, 0.f, 0.f, 0.f, 0.f, 0.f, 0.f, 0.f};
    v8f acc1 = {0.f, 0.f, 0.f, 0.f, 0.f, 0.f, 0.f, 0.f};

#pragma unroll 4
    for (int k = 0; k < C_ / 4; ++k) {
        v2f aF = *(const v2f*)(ap + 4 * k);     // w1[o, 4k + 2*half + {0,1}]
        v2f b0, b1f;
        b0.x  = bp0[(4 * k + 0) * ATW];
        b0.y  = bp0[(4 * k + 1) * ATW];
        b1f.x = bp1[(4 * k + 0) * ATW];
        b1f.y = bp1[(4 * k + 1) * ATW];
        acc0 = __builtin_amdgcn_wmma_f32_16x16x4_f32(
            false, aF, false, b0, (short)0, acc0, false, false);
        acc1 = __builtin_amdgcn_wmma_f32_16x16x4_f32(
            false, aF, false, b1f, (short)0, acc1, false, false);
    }

    // ---- Epilogue: depthwise 3/5 from LDS, combine, residual, store (2 tiles) ----
#pragma unroll
    for (int r = 0; r < 8; ++r) {
        // C/D layout: VGPR r -> M = r (lanes 0-15) or r+8 (lanes 16-31); N = lane%16
        const int ol = 16 * w + r + 8 * half;   // 0..127 within block
        const int oA = ob0 + ol;                // absolute channel
        const float k30 = cw3[3 * ol + 0], k31 = cw3[3 * ol + 1], k32 = cw3[3 * ol + 2];
        const float k50 = cw5[5 * ol + 0], k51 = cw5[5 * ol + 1], k52 = cw5[5 * ol + 2];
        const float k53 = cw5[5 * ol + 3], k54 = cw5[5 * ol + 4];
        const float kc0 = cwc[3 * ol + 0], kc1 = cwc[3 * ol + 1], kc2 = cwc[3 * ol + 2];
        const float vb1 = cb1[ol], vb3 = cb3[ol], vb5 = cb5[ol], vbc = cbc[ol];

#pragma unroll
        for (int j = 0; j < 2; ++j) {
            const int tl = 16 * j + l15;              // block-local t
            const float* aw = &aT[oA * ATW + tl];     // window starts at t-2
            const float a0 = aw[0], a1 = aw[1], a2 = aw[2], a3 = aw[3], a4 = aw[4];

            const float g  = (j == 0) ? acc0[r] : acc1[r];
            const float o1 = g + vb1;
            const float o3 = k30 * a1 + k31 * a2 + k32 * a3 + vb3;
            const float o5 = k50 * a0 + k51 * a1 + k52 * a2 + k53 * a3 + k54 * a4 + vb5;

            const size_t gi = ((size_t)bb * C_ + oA) * T_ + (tt0 + tl);
            out[gi] = kc0 * o1 + kc1 * o3 + kc2 * o5 + vbc + x[gi];
        }
    }
}

extern "C" void kernel_launch(void* const* d_in, const int* in_sizes, int n_in,
                              void* d_out, int out_size, void* d_ws, size_t ws_size,
                              hipStream_t stream) {
    (void)in_sizes; (void)n_in; (void)d_ws; (void)ws_size; (void)out_size;
    const float* x  = (const float*)d_in[0];
    const float* w1 = (const float*)d_in[1];
    const float* b1 = (const float*)d_in[2];
    const float* w3 = (const float*)d_in[3];
    const float* b3 = (const float*)d_in[4];
    const float* w5 = (const float*)d_in[5];
    const float* b5 = (const float*)d_in[6];
    const float* wc = (const float*)d_in[7];
    const float* bc = (const float*)d_in[8];
    float* out = (float*)d_out;

    dim3 grid(T_ / TB, C_ / OB, B_);   // (128, 2, 32)
    dim3 block(256);                   // 8 wave32s
    timeconv_inception_kernel<<<grid, block, 0, stream>>>(
        x, w1, b1, w3, b3, w5, b5, wc, bc, out);
}